// GeometryAwareCostVolume_20109036880653
// MI455X (gfx1250) — compile-verified
//
#include <hip/hip_runtime.h>

typedef float v2f __attribute__((ext_vector_type(2)));
typedef float v8f __attribute__((ext_vector_type(8)));

#define GC    8      // channels per group
#define NG    8      // number of groups
#define HH    80
#define WW    160
#define NB    2
#define KT    9      // 2*RADIUS+1
#define NLVL  4
#define NCH   576    // 2*NLVL*NG*KT output channels

#define S0    164    // cv0 LDS row stride (pads so M and M+8 halves hit disjoint banks)
#define S1    84
#define S2    44
#define S3    24
#define FS    164

__global__ __launch_bounds__(256, 1)
void gacv_fused_kernel(const float* __restrict__ fmap1,
                       const float* __restrict__ fmap2,
                       const float* __restrict__ coords,
                       float* __restrict__ out)
{
    // ---- LDS: full (b,g,h) cost-volume slice + pyramid (fits in 320KB WGP LDS) ----
    __shared__ float f1s[GC][FS];      //  5.2 KB
    __shared__ float f2s[GC][FS];      //  5.2 KB
    __shared__ float cv0[WW][S0];      // 105.0 KB  level 0 (W=160)
    __shared__ float lv1[WW][S1];      //  53.8 KB  level 1 (W=80)
    __shared__ float lv2[WW][S2];      //  28.2 KB  level 2 (W=40)
    __shared__ float lv3[WW][S3];      //  15.4 KB  level 3 (W=20)
    __shared__ float cr[WW];           //  0.6 KB

    const int tid = threadIdx.x;
    const int bid = blockIdx.x;              // b*NG*HH + g*HH + h
    const int b   = bid / (NG * HH);
    const int g   = (bid / HH) % NG;
    const int h   = bid % HH;

    // ---- stage 1: load fmap rows (8 channels x 160) and coords row ----
    {
        const size_t base = ((size_t)(b * (NG * GC) + g * GC) * HH + h) * WW;
        for (int idx = tid; idx < GC * WW; idx += 256) {
            const int c = idx / WW, w = idx % WW;
            const size_t off = base + (size_t)c * (HH * WW) + w;
            f1s[c][w] = fmap1[off];
            f2s[c][w] = fmap2[off];
        }
        if (tid < WW) cr[tid] = coords[(size_t)b * HH * WW + (size_t)h * WW + tid];
    }
    __syncthreads();

    // ---- stage 2: cost volume via fp32 WMMA ----
    // cv0[w][v] = (1/sqrt(8)) * sum_c f1s[c][w] * f2s[c][v]
    // D(16x16) = A(16x4)*B(4x16)+C, 10x10 tiles, K=8 -> two k-steps.
    {
        const int wave = tid >> 5;
        const int lane = tid & 31;
        const int l    = lane & 15;
        const int hi   = lane >> 4;          // 0: K={0,1}/M=r half, 1: K={2,3}/M=r+8 half
        for (int t = wave; t < 100; t += 8) {
            const int m0 = (t / 10) * 16;
            const int n0 = (t % 10) * 16;
            v8f acc = {};
#pragma unroll
            for (int s = 0; s < 2; ++s) {
                v2f av, bv;
                // A[M=m0+l, K=s*4+hi*2+{0,1}] = f1^T -> f1s[c][m]
                av.x = f1s[s * 4 + hi * 2 + 0][m0 + l];
                av.y = f1s[s * 4 + hi * 2 + 1][m0 + l];
                // B[K=s*4+hi*2+{0,1}, N=n0+l] = f2s[c][v]
                bv.x = f2s[s * 4 + hi * 2 + 0][n0 + l];
                bv.y = f2s[s * 4 + hi * 2 + 1][n0 + l];
                acc = __builtin_amdgcn_wmma_f32_16x16x4_f32(
                    false, av, false, bv, (short)0, acc, false, false);
            }
#pragma unroll
            for (int r = 0; r < 8; ++r) {
                cv0[m0 + hi * 8 + r][n0 + l] = acc[r] * 0.35355339059327373f;
            }
        }
    }
    __syncthreads();

    // ---- stage 3: pyramid (1-D avg pool k=2 s=2 along v) ----
    for (int idx = tid; idx < WW * 80; idx += 256) {
        const int r = idx / 80, c = idx % 80;
        lv1[r][c] = 0.5f * (cv0[r][2 * c] + cv0[r][2 * c + 1]);
    }
    __syncthreads();
    for (int idx = tid; idx < WW * 40; idx += 256) {
        const int r = idx / 40, c = idx % 40;
        lv2[r][c] = 0.5f * (lv1[r][2 * c] + lv1[r][2 * c + 1]);
    }
    __syncthreads();
    for (int idx = tid; idx < WW * 20; idx += 256) {
        const int r = idx / 20, c = idx % 20;
        lv3[r][c] = 0.5f * (lv2[r][2 * c] + lv2[r][2 * c + 1]);
    }
    __syncthreads();

    // ---- stage 4: 1-D bilinear sampling (9 taps, 4 levels) + coalesced writes ----
    const size_t HW = (size_t)HH * WW;
    for (int idx = tid; idx < WW * NLVL; idx += 256) {
        const int w   = idx % WW;
        const int lvl = idx / WW;
        const float x = cr[w] / (float)(1 << lvl);
        const float* row;
        int Wl;
        if      (lvl == 0) { row = &cv0[w][0]; Wl = 160; }
        else if (lvl == 1) { row = &lv1[w][0]; Wl = 80;  }
        else if (lvl == 2) { row = &lv2[w][0]; Wl = 40;  }
        else               { row = &lv3[w][0]; Wl = 20;  }
#pragma unroll
        for (int k = 0; k < KT; ++k) {
            const float pos = x + (float)(k - 4);
            const float x0  = floorf(pos);
            const float fr  = pos - x0;
            const int i0 = (int)x0;
            const int i1 = i0 + 1;
            const int c0 = i0 < 0 ? 0 : (i0 > Wl - 1 ? Wl - 1 : i0);
            const int c1 = i1 < 0 ? 0 : (i1 > Wl - 1 ? Wl - 1 : i1);
            const float v0 = (i0 >= 0 && i0 < Wl) ? row[c0] * (1.0f - fr) : 0.0f;
            const float v1 = (i1 >= 0 && i1 < Wl) ? row[c1] * fr         : 0.0f;
            const float val = v0 + v1;
            const int ch = ((lvl * 2) * NG + g) * KT + k;           // p=0 (feat)
            const size_t o = ((size_t)b * NCH + ch) * HW + (size_t)h * WW + w;
            out[o] = val;
            out[o + (size_t)(NG * KT) * HW] = val;                  // p=1 (geo == feat)
        }
    }
}

extern "C" void kernel_launch(void* const* d_in, const int* in_sizes, int n_in,
                              void* d_out, int out_size, void* d_ws, size_t ws_size,
                              hipStream_t stream) {
    const float* fmap1  = (const float*)d_in[0];
    const float* fmap2  = (const float*)d_in[1];
    const float* coords = (const float*)d_in[2];
    float* out = (float*)d_out;

    dim3 grid(NB * NG * HH);   // 1280 workgroups, one (b,g,h) slice each
    dim3 block(256);           // 8 wave32s
    gacv_fused_kernel<<<grid, block, 0, stream>>>(fmap1, fmap2, coords, out);
}